// PointSamplingNetMSG_14637248545010
// MI455X (gfx1250) — compile-verified
//
#include <hip/hip_runtime.h>
#include <hip/hip_bf16.h>

// PointSamplingNetMSG for MI455X (gfx1250, wave32).
//   K1: per-point MLP 3->32->64->256 (BN folded, ReLU) -> X as k-pair float2 [B][128][M]
//   K2: head GEMM Q = sigmoid(head_w * X + b) via V_WMMA_F32_16X16X4_F32 (f32 kept for
//       exact ranking), B-frags as single global_load_b64 thanks to the paired layout
//   K3: per (b,s) row exact tie-stable top-64 (incremental local max + wave shuffle
//       reduce), then float4 gathers into the 6 outputs.
// Workspace: X = 128 MiB + Q group slot (4 batches) = 64 MiB -> 192 MiB total.

#define Bn 16
#define Mn 8192
#define Sn 512
#define Cn 64
#define GRP 4   // batches per Q staging group

typedef __attribute__((ext_vector_type(2))) float v2f;
typedef __attribute__((ext_vector_type(8))) float v8f;
typedef unsigned long long ull;

// ---------------------------------------------------------------------------
// Kernel 1: per-point MLP. One thread per point. Grid = B*M/256.
// Output layout: X2[b][kk][m] = float2{ x[2kk][m], x[2kk+1][m] }, kk in 0..127.
// ---------------------------------------------------------------------------
__global__ __launch_bounds__(256) void psn_mlp_kernel(
    const float* __restrict__ coord,
    const float* __restrict__ w0, const float* __restrict__ b0,
    const float* __restrict__ g0, const float* __restrict__ be0,
    const float* __restrict__ rm0, const float* __restrict__ rv0,
    const float* __restrict__ w1, const float* __restrict__ b1,
    const float* __restrict__ g1, const float* __restrict__ be1,
    const float* __restrict__ rm1, const float* __restrict__ rv1,
    const float* __restrict__ w2, const float* __restrict__ b2,
    const float* __restrict__ g2, const float* __restrict__ be2,
    const float* __restrict__ rm2, const float* __restrict__ rv2,
    float2* __restrict__ X2)  // [B][128][M] float2
{
  const int gid = blockIdx.x * 256 + threadIdx.x;   // gid = b*M + m
  const int b = gid >> 13;
  const int m = gid & (Mn - 1);

  const float c0 = coord[gid * 3 + 0];
  const float c1 = coord[gid * 3 + 1];
  const float c2 = coord[gid * 3 + 2];

  // Layer 0: 3 -> 32
  float h1[32];
#pragma unroll
  for (int o = 0; o < 32; ++o) {
    float acc = w0[o * 3 + 0] * c0 + w0[o * 3 + 1] * c1 + w0[o * 3 + 2] * c2;
    const float s = g0[o] * __frsqrt_rn(rv0[o] + 1e-5f);
    acc = acc * s + ((b0[o] - rm0[o]) * s + be0[o]);
    h1[o] = fmaxf(acc, 0.0f);
  }

  // Layer 1: 32 -> 64
  float h2[64];
#pragma unroll
  for (int o = 0; o < 64; ++o) {
    float acc = 0.0f;
#pragma unroll
    for (int i = 0; i < 32; ++i) acc += w1[o * 32 + i] * h1[i];
    const float s = g1[o] * __frsqrt_rn(rv1[o] + 1e-5f);
    acc = acc * s + ((b1[o] - rm1[o]) * s + be1[o]);
    h2[o] = fmaxf(acc, 0.0f);
  }

  // Layer 2: 64 -> 256, outputs in channel pairs -> coalesced b64 stores
  float2* __restrict__ Xb = X2 + (size_t)b * 128 * Mn + m;
  for (int oo = 0; oo < 128; ++oo) {
    float a0 = 0.0f, a1 = 0.0f;
#pragma unroll
    for (int i = 0; i < 64; ++i) {
      a0 += w2[(2 * oo + 0) * 64 + i] * h2[i];
      a1 += w2[(2 * oo + 1) * 64 + i] * h2[i];
    }
    const int o0 = 2 * oo, o1 = 2 * oo + 1;
    const float s0 = g2[o0] * __frsqrt_rn(rv2[o0] + 1e-5f);
    const float s1 = g2[o1] * __frsqrt_rn(rv2[o1] + 1e-5f);
    a0 = a0 * s0 + ((b2[o0] - rm2[o0]) * s0 + be2[o0]);
    a1 = a1 * s1 + ((b2[o1] - rm2[o1]) * s1 + be2[o1]);
    Xb[(size_t)oo * Mn] = make_float2(fmaxf(a0, 0.0f), fmaxf(a1, 0.0f));
  }
}

// ---------------------------------------------------------------------------
// Kernel 2 (per batch group): Q = sigmoid(head_w * X + head_b), f32 WMMA.
// Grid: (S/16 s-tiles, M/512 m-chunks, GRP batches). 256 thr = 8 waves,
// each wave owns 4 m-tiles of 16; K-loop: 64 x V_WMMA_F32_16X16X4_F32.
// ---------------------------------------------------------------------------
__global__ __launch_bounds__(256) void psn_head_kernel(
    const float2* __restrict__ X2,     // [B][128][M] float2
    const float* __restrict__ head_w,  // [512][256]
    const float* __restrict__ head_b,  // [512]
    float* __restrict__ Qg,            // [GRP][512][M]
    const int b0)
{
  __shared__ float wlds[16 * 256];
  __shared__ float blds[16];

  const int s0 = blockIdx.x * 16;       // s-tile base
  const int mc = blockIdx.y * 512;      // m-chunk base
  const int bl = blockIdx.z;            // batch within group
  const int tid = threadIdx.x;

  const float2* __restrict__ Xb = X2 + (size_t)(b0 + bl) * 128 * Mn;
  float* __restrict__ Q = Qg + (size_t)bl * Sn * Mn;

  for (int i = tid; i < 16 * 256; i += 256) wlds[i] = head_w[s0 * 256 + i];
  if (tid < 16) blds[tid] = head_b[s0 + tid];
  __syncthreads();

  const int wave = tid >> 5;
  const int lane = tid & 31;
  const int hl = lane >> 4;     // half-wave select (K interleave)
  const int ln = lane & 15;     // row-of-A / col-of-B index

  const float* __restrict__ wrow = wlds + ln * 256 + 2 * hl;

  for (int t = 0; t < 4; ++t) {
    const int m0 = mc + (wave * 4 + t) * 16;
    const float2* __restrict__ xcol = Xb + m0 + ln;
    v8f c = {};
#pragma unroll 4
    for (int j = 0; j < 64; ++j) {
      // A frag (16x4, lane=row s, a[i] -> K = 4j + 2*hl + i)   [ISA 7.12.2]
      v2f a;
      a[0] = wrow[4 * j + 0];
      a[1] = wrow[4 * j + 1];
      // B frag (4x16, lane=col m): one b64 load, K pair contiguous by layout
      const float2 t2 = xcol[(size_t)(2 * j + hl) * Mn];
      v2f bf;
      bf[0] = t2.x;
      bf[1] = t2.y;
      c = __builtin_amdgcn_wmma_f32_16x16x4_f32(
          /*neg_a=*/false, a, /*neg_b=*/false, bf,
          /*c_mod=*/(short)0, c, /*reuse_a=*/false, /*reuse_b=*/false);
    }
    // C layout: VGPR v -> row s = v + 8*hl, col = ln
#pragma unroll
    for (int v = 0; v < 8; ++v) {
      const int srow = v + 8 * hl;
      const float x = c[v] + blds[srow];
      const float q = __builtin_amdgcn_rcpf(1.0f + __expf(-x));  // sigmoid
      Q[(size_t)(s0 + srow) * Mn + m0 + ln] = q;
    }
  }
}

// ---------------------------------------------------------------------------
// Kernel 3 (per batch group): exact tie-stable top-64 per (b,s) row + gather.
// One block (256 thr) per row; grid (S, GRP). Keys in registers (32/thread).
// key = (float_bits(q) << 32) | (8192 - m): q>0 so bits order-preserve; ties
// resolve to the smaller m (matches stable argsort of -Q). Incremental local
// max: only the round winner rescans; wave shuffle reduce, 2 barriers/round.
// ---------------------------------------------------------------------------
__global__ __launch_bounds__(256) void psn_topk_gather_kernel(
    const float* __restrict__ Qg,     // [GRP][S][M]
    const float* __restrict__ coord,  // [B][M][3]
    const float* __restrict__ feat,   // [B][M][64]
    float* __restrict__ out, const int b0)
{
  __shared__ ull red[8];
  __shared__ int topm[64];

  const int s = blockIdx.x;
  const int bl = blockIdx.y;
  const int b = b0 + bl;
  const int tid = threadIdx.x;
  const int wave = tid >> 5;
  const int lane = tid & 31;

  const float4* __restrict__ qrow4 =
      (const float4*)(Qg + ((size_t)bl * Sn + s) * Mn);

  ull key[32];
#pragma unroll
  for (int i = 0; i < 8; ++i) {
    const int e4 = tid + 256 * i;
    const float4 f = qrow4[e4];
    const int mb = 4 * e4;
    key[4 * i + 0] = ((ull)__float_as_uint(f.x) << 32) | (unsigned)(Mn - (mb + 0));
    key[4 * i + 1] = ((ull)__float_as_uint(f.y) << 32) | (unsigned)(Mn - (mb + 1));
    key[4 * i + 2] = ((ull)__float_as_uint(f.z) << 32) | (unsigned)(Mn - (mb + 2));
    key[4 * i + 3] = ((ull)__float_as_uint(f.w) << 32) | (unsigned)(Mn - (mb + 3));
  }

  unsigned taken = 0u;
  ull lmax = 0ull;
  int larg = 0;
#pragma unroll
  for (int i = 0; i < 32; ++i)
    if (key[i] > lmax) { lmax = key[i]; larg = i; }

  for (int r = 0; r < 64; ++r) {
    // wave-level max of cached local maxima (no barriers)
    ull wmax = lmax;
#pragma unroll
    for (int d = 16; d >= 1; d >>= 1) {
      const ull o = __shfl_xor(wmax, d, 32);
      if (o > wmax) wmax = o;
    }
    if (lane == 0) red[wave] = wmax;
    __syncthreads();
    ull g = red[0];
#pragma unroll
    for (int w = 1; w < 8; ++w)
      if (red[w] > g) g = red[w];
    if (tid == 0) topm[r] = Mn - (int)(unsigned)g;
    if (lmax == g) {                 // unique winner (keys are unique)
      taken |= (1u << larg);
      lmax = 0ull; larg = 0;
#pragma unroll
      for (int i = 0; i < 32; ++i)
        if (!((taken >> i) & 1u) && key[i] > lmax) { lmax = key[i]; larg = i; }
    }
    __syncthreads();                 // red[] reuse + topm publication
  }

  // ---- gather ----
  const long bs = (long)b * Sn + s;
  float* __restrict__ o_sp   = out;                    // [B,S,3]
  float* __restrict__ o_gp32 = out + 24576;            // [B,S,32,3]
  float* __restrict__ o_gp64 = o_gp32 + 786432;        // [B,S,64,3]
  float* __restrict__ o_sf   = o_gp64 + 1572864;       // [B,S,64]
  float* __restrict__ o_gf32 = o_sf + 524288;          // [B,S,32,64]
  float* __restrict__ o_gf64 = o_gf32 + 16777216;      // [B,S,64,64]

  for (int e = tid; e < 64 * 3; e += 256) {
    const int j = e / 3, cc = e - 3 * j;
    const float v = coord[((long)b * Mn + topm[j]) * 3 + cc];
    o_gp64[(bs * 64 + j) * 3 + cc] = v;
    if (j < 32) o_gp32[(bs * 32 + j) * 3 + cc] = v;
    if (j == 0) o_sp[bs * 3 + cc] = v;
  }
  const float4* __restrict__ feat4 = (const float4*)(feat + (long)b * Mn * 64);
  float4* __restrict__ gf64_4 = (float4*)o_gf64;
  float4* __restrict__ gf32_4 = (float4*)o_gf32;
  float4* __restrict__ sf_4   = (float4*)o_sf;
  for (int e = tid; e < 64 * 16; e += 256) {  // 16 float4 per gathered row
    const int j = e >> 4, c4 = e & 15;
    const float4 v = feat4[(size_t)topm[j] * 16 + c4];
    gf64_4[(bs * 64 + j) * 16 + c4] = v;
    if (j < 32) gf32_4[(bs * 32 + j) * 16 + c4] = v;
    if (j == 0) sf_4[bs * 16 + c4] = v;
  }
}

// ---------------------------------------------------------------------------
extern "C" void kernel_launch(void* const* d_in, const int* in_sizes, int n_in,
                              void* d_out, int out_size, void* d_ws, size_t ws_size,
                              hipStream_t stream) {
  const float* coord = (const float*)d_in[0];
  const float* feat  = (const float*)d_in[1];
  const float *cw[3], *cb[3], *bg[3], *bb[3], *bm[3], *bv[3];
  for (int i = 0; i < 3; ++i) {
    cw[i] = (const float*)d_in[2 + 6 * i + 0];
    cb[i] = (const float*)d_in[2 + 6 * i + 1];
    bg[i] = (const float*)d_in[2 + 6 * i + 2];
    bb[i] = (const float*)d_in[2 + 6 * i + 3];
    bm[i] = (const float*)d_in[2 + 6 * i + 4];
    bv[i] = (const float*)d_in[2 + 6 * i + 5];
  }
  const float* head_w = (const float*)d_in[20];
  const float* head_b = (const float*)d_in[21];

  float2* X2 = (float2*)d_ws;                        // B*128*M float2 = 128 MiB
  float* Qg  = (float*)d_ws + (size_t)Bn * 256 * Mn; // GRP*S*M f32    =  64 MiB
  float* out = (float*)d_out;

  psn_mlp_kernel<<<dim3((Bn * Mn) / 256), 256, 0, stream>>>(
      coord,
      cw[0], cb[0], bg[0], bb[0], bm[0], bv[0],
      cw[1], cb[1], bg[1], bb[1], bm[1], bv[1],
      cw[2], cb[2], bg[2], bb[2], bm[2], bv[2],
      X2);

  for (int g = 0; g < Bn / GRP; ++g) {
    psn_head_kernel<<<dim3(Sn / 16, Mn / 512, GRP), 256, 0, stream>>>(
        X2, head_w, head_b, Qg, g * GRP);
    psn_topk_gather_kernel<<<dim3(Sn, GRP), 256, 0, stream>>>(
        Qg, coord, feat, out, g * GRP);
  }
}